// NGCN_1056561954826
// MI455X (gfx1250) — compile-verified
//
#include <hip/hip_runtime.h>

// ---------------------------------------------------------------------------
// NGCN on MI455X (gfx1250): 2x [Linear -> L2norm*1.8 -> GraphConv(norm=both)]
// Strategy:
//  * Build CSR over dst once (degree atomics -> single-block scan -> bucket).
//  * GEMM via V_WMMA_F32_16X16X4_F32 (true fp32 matrix op), fused with bias,
//    row L2-normalization, SCALE and D_out^{-1/2} pre-scaling.
//  * Aggregation = CSR gather (L2-resident: h is 51.2MB << 192MB L2),
//    one wave32 per node, float4 coalesced loads + global prefetch.
// ---------------------------------------------------------------------------

typedef __attribute__((ext_vector_type(2))) float v2f;
typedef __attribute__((ext_vector_type(8))) float v8f;

#define D_FEAT 128
#define SCALE_F 1.8f
#define EPS_F 1e-12f

// --------------------------- degree counting -------------------------------
__global__ void __launch_bounds__(256) degree_kernel(
    const int* __restrict__ src, const int* __restrict__ dst,
    int* __restrict__ deg_out, int* __restrict__ deg_in, int E) {
  int i = blockIdx.x * blockDim.x + threadIdx.x;
  if (i < E) {
    atomicAdd(&deg_out[src[i]], 1);
    atomicAdd(&deg_in[dst[i]], 1);
  }
}

// --------------------------- degree -> rsqrt norms -------------------------
__global__ void __launch_bounds__(256) norm_kernel(
    const int* __restrict__ deg_out, const int* __restrict__ deg_in,
    float* __restrict__ norm_out, float* __restrict__ norm_in, int N) {
  int i = blockIdx.x * blockDim.x + threadIdx.x;
  if (i < N) {
    int dn = deg_out[i]; if (dn < 1) dn = 1;
    int di = deg_in[i];  if (di < 1) di = 1;
    norm_out[i] = rsqrtf((float)dn);
    norm_in[i]  = rsqrtf((float)di);
  }
}

// ------------------- single-block exclusive scan (CSR offsets) -------------
__global__ void __launch_bounds__(1024) scan_kernel(
    const int* __restrict__ counts, int* __restrict__ offsets,
    int* __restrict__ cursor, int N) {
  __shared__ int sdata[1024];
  __shared__ int s_carry;
  if (threadIdx.x == 0) s_carry = 0;
  __syncthreads();
  for (int base = 0; base < N; base += 1024) {
    int i = base + (int)threadIdx.x;
    int v = (i < N) ? counts[i] : 0;
    sdata[threadIdx.x] = v;
    __syncthreads();
    for (int off = 1; off < 1024; off <<= 1) {
      int t = (threadIdx.x >= (unsigned)off) ? sdata[threadIdx.x - off] : 0;
      __syncthreads();
      sdata[threadIdx.x] += t;
      __syncthreads();
    }
    int incl  = sdata[threadIdx.x];
    int total = sdata[1023];
    int carry = s_carry;
    if (i < N) {
      int o = carry + incl - v;  // exclusive prefix
      offsets[i] = o;
      cursor[i]  = o;
    }
    __syncthreads();
    if (threadIdx.x == 0) s_carry = carry + total;
    __syncthreads();
  }
  if (threadIdx.x == 0) offsets[N] = s_carry;
}

// ------------------- bucket edges by dst (CSR column array) ----------------
__global__ void __launch_bounds__(256) bucket_kernel(
    const int* __restrict__ src, const int* __restrict__ dst,
    int* __restrict__ cursor, int* __restrict__ esrc, int E) {
  int i = blockIdx.x * blockDim.x + threadIdx.x;
  if (i < E) {
    int p = atomicAdd(&cursor[dst[i]], 1);
    esrc[p] = src[i];
  }
}

// ---------------------------------------------------------------------------
// Fused: hpre = (X @ W^T + b), per-row L2-normalized * 1.8 * norm_out[row].
// One wave32 per 16-row strip, 8 column tiles of 16 (full 128 cols in regs).
// fp32 WMMA 16x16x4: per ISA layouts:
//   A 16x4 : lane l (m=l&15, half=l>>4): vgpr0=K(2*half), vgpr1=K(2*half+1)
//   B 4x16 : lane l (n=l&15, half=l>>4): vgpr0=K(2*half), vgpr1=K(2*half+1)
//   C/D    : lane l, vgpr v -> row m = v + 8*half, col n = l&15
// ---------------------------------------------------------------------------
__global__ void __launch_bounds__(256) gemm_norm_kernel(
    const float* __restrict__ X, const float* __restrict__ W,
    const float* __restrict__ bias, const float* __restrict__ norm_out,
    float* __restrict__ hpre, int nstrips) {
  int wid  = blockIdx.x * (blockDim.x >> 5) + ((int)threadIdx.x >> 5);
  if (wid >= nstrips) return;  // wave-granular: surviving waves have EXEC all-1s
  int lane = (int)threadIdx.x & 31;
  int half = lane >> 4;        // 0 or 1
  int mn   = lane & 15;        // A-row (m) and B/C/D-col (n) for this lane
  int row0 = wid * 16;
  int kb   = 2 * half;

  v8f acc[8];
#pragma unroll
  for (int t = 0; t < 8; ++t) {
    v8f z = {0.f, 0.f, 0.f, 0.f, 0.f, 0.f, 0.f, 0.f};
    acc[t] = z;
  }

  const float* xrow = X + (size_t)(row0 + mn) * D_FEAT;

  for (int k = 0; k < D_FEAT; k += 4) {
    float2 a2 = *(const float2*)(xrow + k + kb);
    v2f a; a.x = a2.x; a.y = a2.y;
#pragma unroll
    for (int t = 0; t < 8; ++t) {
      // B(k',n) = W^T[k'][n] = W[16t+n][k']   (torch Linear: h @ W.T)
      float2 b2 = *(const float2*)(W + (size_t)(t * 16 + mn) * D_FEAT + k + kb);
      v2f b; b.x = b2.x; b.y = b2.y;
      acc[t] = __builtin_amdgcn_wmma_f32_16x16x4_f32(
          false, a, false, b, (short)0, acc[t], false, false);
    }
  }

  // bias add (col-indexed: same for all 8 row-vgprs of a tile)
#pragma unroll
  for (int t = 0; t < 8; ++t) {
    float bv = bias[t * 16 + mn];
#pragma unroll
    for (int v = 0; v < 8; ++v) acc[t][v] += bv;
  }

  // per-row sum of squares -> normalize -> *SCALE -> *norm_out[row] -> store
#pragma unroll
  for (int v = 0; v < 8; ++v) {
    float s = 0.f;
#pragma unroll
    for (int t = 0; t < 8; ++t) s += acc[t][v] * acc[t][v];
    // reduce over the 16 lanes of this half (bit4 of lane untouched)
    s += __shfl_xor(s, 1, 32);
    s += __shfl_xor(s, 2, 32);
    s += __shfl_xor(s, 4, 32);
    s += __shfl_xor(s, 8, 32);
    int row  = row0 + v + 8 * half;
    float nr = sqrtf(s);
    float f  = (SCALE_F / fmaxf(nr, EPS_F)) * norm_out[row];
    float* orow = hpre + (size_t)row * D_FEAT;
#pragma unroll
    for (int t = 0; t < 8; ++t) orow[t * 16 + mn] = acc[t][v] * f;
  }
}

// ---------------------------------------------------------------------------
// CSR gather-aggregate: out[n] = norm_in[n] * sum_{e: dst==n} hpre[src(e)].
// One wave32 per node; 4 floats/lane -> 512B coalesced row loads (L2-hot).
// ---------------------------------------------------------------------------
__global__ void __launch_bounds__(256) agg_kernel(
    const float* __restrict__ hpre, const int* __restrict__ esrc,
    const int* __restrict__ offsets, const float* __restrict__ norm_in,
    float* __restrict__ out, int N) {
  int wid  = blockIdx.x * (blockDim.x >> 5) + ((int)threadIdx.x >> 5);
  int lane = (int)threadIdx.x & 31;
  if (wid >= N) return;
  int j0 = offsets[wid], j1 = offsets[wid + 1];
  float ax = 0.f, ay = 0.f, az = 0.f, aw = 0.f;
  for (int j = j0; j < j1; ++j) {
    int s = esrc[j];
    if (j + 1 < j1)  // hide latency of the next random row
      __builtin_prefetch(hpre + (size_t)esrc[j + 1] * D_FEAT + lane * 4, 0, 3);
    float4 v = ((const float4*)(hpre + (size_t)s * D_FEAT))[lane];
    ax += v.x; ay += v.y; az += v.z; aw += v.w;
  }
  float ni = norm_in[wid];
  float4 r; r.x = ax * ni; r.y = ay * ni; r.z = az * ni; r.w = aw * ni;
  ((float4*)(out + (size_t)wid * D_FEAT))[lane] = r;
}

// ---------------------------------------------------------------------------
extern "C" void kernel_launch(void* const* d_in, const int* in_sizes, int n_in,
                              void* d_out, int out_size, void* d_ws, size_t ws_size,
                              hipStream_t stream) {
  const float* x  = (const float*)d_in[0];
  const float* W1 = (const float*)d_in[1];
  const float* b1 = (const float*)d_in[2];
  const float* W2 = (const float*)d_in[3];
  const float* b2 = (const float*)d_in[4];
  const int*   src = (const int*)d_in[5];
  const int*   dst = (const int*)d_in[6];

  const int N = in_sizes[0] / D_FEAT;   // 100000
  const int E = in_sizes[5];            // 1600000

  // workspace carve-up (256B aligned): ~60 MB total
  char* w = (char*)d_ws;
  auto carve = [&](size_t bytes) -> void* {
    void* p = (void*)w;
    w += (bytes + 255) & ~(size_t)255;
    return p;
  };
  float* hpre     = (float*)carve((size_t)N * D_FEAT * sizeof(float));
  float* norm_out = (float*)carve((size_t)N * sizeof(float));
  float* norm_in  = (float*)carve((size_t)N * sizeof(float));
  int*   dego     = (int*)carve((size_t)N * sizeof(int));
  int*   degi     = (int*)carve((size_t)N * sizeof(int));
  int*   offsets  = (int*)carve((size_t)(N + 1) * sizeof(int));
  int*   cursor   = (int*)carve((size_t)N * sizeof(int));
  int*   esrc     = (int*)carve((size_t)E * sizeof(int));

  const int TB = 256;
  hipMemsetAsync(dego, 0, (size_t)N * sizeof(int), stream);
  hipMemsetAsync(degi, 0, (size_t)N * sizeof(int), stream);

  degree_kernel<<<(E + TB - 1) / TB, TB, 0, stream>>>(src, dst, dego, degi, E);
  norm_kernel<<<(N + TB - 1) / TB, TB, 0, stream>>>(dego, degi, norm_out, norm_in, N);
  scan_kernel<<<1, 1024, 0, stream>>>(degi, offsets, cursor, N);
  bucket_kernel<<<(E + TB - 1) / TB, TB, 0, stream>>>(src, dst, cursor, esrc, E);

  const int strips = N / 16;            // N = 100000 is divisible by 16
  const int wpb = TB / 32;              // 8 waves per block
  const int gblocks = (strips + wpb - 1) / wpb;
  const int ablocks = (N + wpb - 1) / wpb;

  // Layer 1: x -> hpre -> aggregate into d_out (doubles as h1)
  gemm_norm_kernel<<<gblocks, TB, 0, stream>>>(x, W1, b1, norm_out, hpre, strips);
  agg_kernel<<<ablocks, TB, 0, stream>>>(hpre, esrc, offsets, norm_in,
                                         (float*)d_out, N);
  // Layer 2: h1 (in d_out) -> hpre -> aggregate into d_out (final)
  gemm_norm_kernel<<<gblocks, TB, 0, stream>>>((const float*)d_out, W2, b2,
                                               norm_out, hpre, strips);
  agg_kernel<<<ablocks, TB, 0, stream>>>(hpre, esrc, offsets, norm_in,
                                         (float*)d_out, N);
}